// DistMaps_27814208209730
// MI455X (gfx1250) — compile-verified
//
#include <hip/hip_runtime.h>
#include <hip/hip_bf16.h>

typedef __attribute__((ext_vector_type(2))) float v2f;
typedef __attribute__((ext_vector_type(8))) float v8f;

// Problem constants (fixed by the reference)
#define NB    8
#define P2    48
#define HGT   512
#define WID   512
#define HW    (HGT * WID)
#define WSH   9      // log2(WID)
#define WMSK  511

// exponent = k0*x^2 + k1*x + k2*y^2 + k3*y + cinit   (== -d2/r^2)
// Layout per (batch,channel): 24 valid points padded to 32 point-slots.
// Invalid / pad slots: k*=0, cinit=-1e30 -> exp() == 0, no masking needed.

__global__ __launch_bounds__(256) void distmaps_wmma_kernel(
    const float* __restrict__ coords,  // [B, P2, 4] = (px, py, order, rad)
    float* __restrict__ out)           // [B, 2, H, W]
{
    __shared__ float sK0[32], sK1[32], sK2[32], sK3[32], sC[32];

    const int bc = blockIdx.y;          // 0..15
    const int b  = bc >> 1;
    const int ch = bc & 1;

    if (threadIdx.x == 0) {
        // pull the (tiny) coord block toward the caches early
        __builtin_prefetch(coords + ((size_t)b * P2) * 4, 0, 3);
    }

    // ---- one wave builds the per-point WMMA coefficients in LDS ----
    if (threadIdx.x < 32) {
        const int t = threadIdx.x;
        float k0 = 0.0f, k1 = 0.0f, k2 = 0.0f, k3 = 0.0f, ci = -1.0e30f;
        if (t < 24) {
            const float* cp = coords + ((size_t)b * P2 + ch * 24 + t) * 4;
            const float px  = cp[0];
            const float py  = cp[1];
            const float rad = cp[3];
            if (px >= 0.0f && py >= 0.0f) {
                const float ir2 = 1.0f / (rad * rad);
                k0 = -ir2;
                k1 = 2.0f * px * ir2;
                k2 = -ir2;
                k3 = 2.0f * py * ir2;
                ci = -(px * px + py * py) * ir2;
            }
        }
        sK0[t] = k0; sK1[t] = k1; sK2[t] = k2; sK3[t] = k3; sC[t] = ci;
    }
    __syncthreads();

    const int lane = threadIdx.x & 31;
    const int wave = threadIdx.x >> 5;
    const int hi   = lane >> 4;     // 0: lanes 0-15 carry K0/K1 (A) rows M=0..7 (C/D)
    const int nl   = lane & 15;

    // ---- A fragments: 16 points (M) x 4 coeffs (K); two point tiles ----
    // A layout: lanes 0-15 -> {K0,K1}, lanes 16-31 -> {K2,K3}, M = lane&15.
    v2f a0, a1;
    a0.x = hi ? sK2[nl]      : sK0[nl];
    a0.y = hi ? sK3[nl]      : sK1[nl];
    a1.x = hi ? sK2[nl + 16] : sK0[nl + 16];
    a1.y = hi ? sK3[nl + 16] : sK1[nl + 16];

    // ---- C fragments: bias per point, broadcast over the 16 pixel cols ----
    // C/D layout: vgpr r holds row M = r + 8*hi.
    v8f c0, c1;
#pragma unroll
    for (int r = 0; r < 8; ++r) {
        c0[r] = sC[hi * 8 + r];
        c1[r] = sC[16 + hi * 8 + r];
    }

    float* __restrict__ outBC = out + (size_t)bc * HW;
    const int waveBase = blockIdx.x * 1024 + wave * 128;

#pragma unroll
    for (int it = 0; it < 4; ++it) {
        const int p0 = waveBase + it * 32;   // 32 consecutive pixels this iter
        float m01[2];
#pragma unroll
        for (int t = 0; t < 2; ++t) {
            // B fragment: 4 features (K) x 16 pixels (N).
            // lanes 0-15 -> {x^2, x}, lanes 16-31 -> {y^2, y}, N = lane&15.
            const int   pix  = p0 + t * 16 + nl;
            const float rowf = (float)(pix >> WSH);
            const float colf = (float)(pix & WMSK);
            v2f bf;
            bf.x = hi ? colf * colf : rowf * rowf;
            bf.y = hi ? colf        : rowf;

            // D = A x B + C : 16 points x 16 pixels of gaussian exponents
            v8f d0 = __builtin_amdgcn_wmma_f32_16x16x4_f32(
                false, a0, false, bf, (short)0, c0, false, false);
            v8f d1 = __builtin_amdgcn_wmma_f32_16x16x4_f32(
                false, a1, false, bf, (short)0, c1, false, false);

            // max over this lane's 16 point-slots (reference inits max at 0)
            float mx = 0.0f;
#pragma unroll
            for (int r = 0; r < 8; ++r) {
                mx = fmaxf(mx, __expf(d0[r]));
                mx = fmaxf(mx, __expf(d1[r]));
            }
            // combine the two lane halves (points 0-7/16-23 vs 8-15/24-31)
            mx = fmaxf(mx, __shfl_xor(mx, 16, 32));
            m01[t] = mx;
        }
        // lanes 0-15 store tile0's 16 pixels, lanes 16-31 store tile1's:
        // one fully coalesced 128B wave store; output is write-once -> NT.
        const int   idx = p0 + (hi ? 16 : 0) + nl;
        const float val = hi ? m01[1] : m01[0];
        __builtin_nontemporal_store(val, &outBC[idx]);
    }
}

extern "C" void kernel_launch(void* const* d_in, const int* in_sizes, int n_in,
                              void* d_out, int out_size, void* d_ws, size_t ws_size,
                              hipStream_t stream) {
    (void)in_sizes; (void)n_in; (void)out_size; (void)d_ws; (void)ws_size;
    // d_in[0] = x [B,C,H,W] (shape-only, unused), d_in[1] = coords [B,P2,4]
    const float* coords = (const float*)d_in[1];
    float* out = (float*)d_out;

    // grid.x: 256 blocks x 1024 px = 262144 px per (b,ch); grid.y: B*2 maps
    dim3 grid(HW / 1024, NB * 2);
    distmaps_wmma_kernel<<<grid, 256, 0, stream>>>(coords, out);
}